// Dictionary_35536559407425
// MI455X (gfx1250) — compile-verified
//
#include <hip/hip_runtime.h>
#include <math.h>

typedef __attribute__((ext_vector_type(2))) float v2f;
typedef __attribute__((ext_vector_type(8))) float v8f;

#define NB 512          // NUM_BASIS == MOTION_DIM == 512
#define EPS 1e-8f

// ---------------------------------------------------------------------------
// Kernel 1: G = (W + 1e-8)^T (W + 1e-8),  G is 512x512.
// grid (32,32) x 256 threads; thread computes one G[i,j].
// ---------------------------------------------------------------------------
__global__ __launch_bounds__(256) void gram_kernel(const float* __restrict__ W,
                                                   float* __restrict__ G) {
    int j = blockIdx.x * 16 + (threadIdx.x & 15);
    int i = blockIdx.y * 16 + (threadIdx.x >> 4);
    float s = 0.f;
    #pragma unroll 4
    for (int r = 0; r < NB; ++r) {
        float a = W[r * NB + i] + EPS;
        float b = W[r * NB + j] + EPS;
        s = fmaf(a, b, s);
    }
    G[i * NB + j] = s;
}

// ---------------------------------------------------------------------------
// Kernel 2: in-place Cholesky of G (lower triangle -> L, G = L*L^T, R = L^T).
// Single block, 1024 threads, right-looking with flat triangle indexing.
// ---------------------------------------------------------------------------
__global__ __launch_bounds__(1024) void chol_kernel(float* __restrict__ G) {
    const int tid = threadIdx.x;
    const int nth = blockDim.x;
    __shared__ float s_inv;

    for (int k = 0; k < NB; ++k) {
        if (tid == 0) {
            float d = sqrtf(G[k * NB + k]);
            G[k * NB + k] = d;
            s_inv = 1.0f / d;
        }
        __syncthreads();
        float inv = s_inv;
        for (int i = k + 1 + tid; i < NB; i += nth)
            G[i * NB + k] *= inv;
        __syncthreads();
        // trailing lower-triangle update: rows i = k+1..511, cols j = k+1..i
        int n = NB - 1 - k;
        int cnt = n * (n + 1) / 2;
        for (int p = tid; p < cnt; p += nth) {
            int r = (int)((sqrtf(8.0f * (float)p + 1.0f) - 1.0f) * 0.5f);
            while ((r + 1) * (r + 2) / 2 <= p) ++r;
            while (r * (r + 1) / 2 > p) --r;
            int i = k + 1 + r;
            int j = k + 1 + (p - r * (r + 1) / 2);
            G[i * NB + j] = fmaf(-G[i * NB + k], G[j * NB + k], G[i * NB + j]);
        }
        __syncthreads();
    }
}

// ---------------------------------------------------------------------------
// Kernel 3: Q = (W + 1e-8) * R^{-1}, row-wise forward substitution:
//   q[j] = (w[i,j] - sum_{k<j} q[k]*L[j,k]) / L[j,j]
// One wave (32 lanes) per row i; q kept in LDS; 8 waves per block.
// ---------------------------------------------------------------------------
__global__ __launch_bounds__(256) void trsm_kernel(const float* __restrict__ W,
                                                   const float* __restrict__ L,
                                                   float* __restrict__ Q) {
    __shared__ float qsh[8][NB];   // 16 KB
    const int lane = threadIdx.x & 31;
    const int w    = threadIdx.x >> 5;
    const int i    = blockIdx.x * 8 + w;   // row of W / Q

    for (int j = 0; j < NB; ++j) {
        float sum = 0.f;
        const float* Lr = L + j * NB;
        for (int k = lane; k < j; k += 32)
            sum = fmaf(qsh[w][k], Lr[k], sum);
        #pragma unroll
        for (int m = 16; m > 0; m >>= 1)
            sum += __shfl_xor(sum, m, 32);
        if (lane == 0) {
            float qj = (W[i * NB + j] + EPS - sum) / Lr[j];
            qsh[w][j] = qj;
            Q[i * NB + j] = qj;
        }
        __syncthreads();   // all 8 waves iterate j in lockstep; orders LDS RAW
    }
}

// ---------------------------------------------------------------------------
// Kernel 4: out = inp @ Q^T via V_WMMA_F32_16X16X4_F32.
// Block = 256 threads = 8 waves; block tile M=32 x N=512.
//   wave w: mhalf = w>>2 (16-row strip), nq = w&3 (128-col strip, 8 subtiles).
// ISA 32-bit 16x4 A layout: lane L<16 -> (M=L, K=0..1) in v[0..1],
//                           lane L+16 -> (M=L, K=2..3)  => one b64/lane/step.
// B = Q^T has the identical per-lane pattern over rows of Q.
// ---------------------------------------------------------------------------
__global__ __launch_bounds__(256) void wmma_gemm_kernel(const float* __restrict__ inp,
                                                        const float* __restrict__ Q,
                                                        float* __restrict__ out) {
    const int lane  = threadIdx.x & 31;
    const int wave  = threadIdx.x >> 5;
    const int mhalf = wave >> 2;          // 0..1
    const int nq    = wave & 3;           // 0..3
    const int lrow  = lane & 15;
    const int lk    = (lane >> 4) * 2;    // 0 or 2

    const long m0 = (long)blockIdx.x * 32 + mhalf * 16;
    const int  n0 = nq * 128;

    const float* pa = inp + (m0 + lrow) * NB + lk;
    const float* pb[8];
    #pragma unroll
    for (int t = 0; t < 8; ++t)
        pb[t] = Q + (long)(n0 + 16 * t + lrow) * NB + lk;

    v8f acc[8];
    #pragma unroll
    for (int t = 0; t < 8; ++t) acc[t] = (v8f)(0.0f);

    #pragma unroll 2
    for (int k0 = 0; k0 < NB; k0 += 4) {
        v2f a = *(const v2f*)(pa + k0);
        #pragma unroll
        for (int t = 0; t < 8; ++t) {
            v2f b = *(const v2f*)(pb[t] + k0);
            acc[t] = __builtin_amdgcn_wmma_f32_16x16x4_f32(
                false, a, false, b, (short)0, acc[t], false, false);
        }
    }

    // C/D layout: VGPR r -> (M = r + 8*(lane>>4), N = lane&15) within tile.
    float* po = out + (m0 + (lane >> 4) * 8) * NB + n0 + lrow;
    #pragma unroll
    for (int t = 0; t < 8; ++t) {
        #pragma unroll
        for (int r = 0; r < 8; ++r)
            po[(long)r * NB + 16 * t] = acc[t][r];
    }
}

// ---------------------------------------------------------------------------
extern "C" void kernel_launch(void* const* d_in, const int* in_sizes, int n_in,
                              void* d_out, int out_size, void* d_ws, size_t ws_size,
                              hipStream_t stream) {
    const float* inp    = (const float*)d_in[0];   // [B, 512] fp32
    const float* weight = (const float*)d_in[1];   // [512, 512] fp32
    float*       out    = (float*)d_out;           // [B, 512] fp32

    float* G = (float*)d_ws;                       // 512*512 floats (L in lower tri)
    float* Q = G + NB * NB;                        // 512*512 floats

    const int B = in_sizes[0] / NB;                // 131072

    gram_kernel<<<dim3(32, 32), 256, 0, stream>>>(weight, G);
    chol_kernel<<<1, 1024, 0, stream>>>(G);
    trsm_kernel<<<NB / 8, 256, 0, stream>>>(weight, G, Q);
    wmma_gemm_kernel<<<B / 32, 256, 0, stream>>>(inp, Q, out);
}